// TreeStructuredAttention_84928683311476
// MI455X (gfx1250) — compile-verified
//
#include <hip/hip_runtime.h>
#include <hip/hip_bf16.h>

// ---------------------------------------------------------------------------
// TreeStructuredAttention on MI455X (gfx1250, wave32)
//
// Key algebra: reshape(H,-1,HS) is a flat reshape => head h of Q/K/V is the
// contiguous row block [h*2048, (h+1)*2048) of the [32768, 64] row view of the
// projected [2048, 1024] matrix. argmax(log_softmax(S/8 + maskterm)) ==
// argmax(S*0.125 + (1-mask)*(-1e4)) since log_softmax is a per-row monotone
// shift. "v" output is the flat V projection verbatim.
//
// fp32 WMMA (V_WMMA_F32_16X16X4_F32) end-to-end: the kernel is bound by the
// 276 MB one-hot output store (~12us at 23.3 TB/s HBM), not matrix math
// (~21.5 GFLOP), and fp32 keeps the argmax bit-faithful.
//
// Cache policy: the 268 MB sampled output and 8 MB V output are write-once /
// never re-read -> non-temporal stores so they don't evict the L2-resident
// working set (Q/K 16 MB read 8x/128x, mask 16.8 MB read 16x, W 4 MB read
// 128x; all << 192 MB L2). Q/K projection stores stay RT: they are the
// L2 producer->consumer hand-off into the attention kernel.
// ---------------------------------------------------------------------------

typedef float v2f __attribute__((ext_vector_type(2)));
typedef float v4f __attribute__((ext_vector_type(4)));
typedef float v8f __attribute__((ext_vector_type(8)));

#define SEQ  2048
#define HID  1024
#define NHEAD 16
#define HS   64

// ---------------------------------------------------------------------------
// C[2048,1024] = X[2048,1024] @ W[1024,1024] + bias.
// 256 threads = 8 waves; each wave owns a 16x64 output strip (4 WMMA tiles),
// K-loop in steps of 4 with V_WMMA_F32_16X16X4_F32.
// Fragment layouts (ISA 7.12.2, fp32 16x16x4):
//   A: lane L -> row m=L&15, a[v] = A[m][k0 + 2*(L>>4) + v]
//   B: lane L -> col n=L&15, b[v] = B[k0 + 2*(L>>4) + v][n]
//   C: lane L -> col n=L&15, acc[v] = C[v + 8*(L>>4)][n]
// NT=true: output is write-once (V projection) -> non-temporal stores.
// ---------------------------------------------------------------------------
template <bool NT>
__global__ __launch_bounds__(256) void proj_gemm_wmma(
    const float* __restrict__ X, const float* __restrict__ W,
    const float* __restrict__ bias, float* __restrict__ C)
{
    const int lane = threadIdx.x & 31;
    const int wave = threadIdx.x >> 5;
    const int gw   = blockIdx.x * 8 + wave;     // 2048 waves total
    const int i0   = (gw >> 4) * 16;            // 128 row blocks
    const int j0   = (gw & 15) * 64;            // 16 col strips
    const int mn   = lane & 15;
    const int half = lane >> 4;

    v8f acc0 = {}, acc1 = {}, acc2 = {}, acc3 = {};

    const float* xrow = X + (size_t)(i0 + mn) * HID;
    for (int k0 = 0; k0 < HID; k0 += 4) {
        v2f a;
        a.x = xrow[k0 + 2 * half + 0];
        a.y = xrow[k0 + 2 * half + 1];

        const float* w0 = W + (size_t)(k0 + 2 * half) * HID + j0 + mn;
        const float* w1 = w0 + HID;
        v2f b0, b1, b2, b3;
        b0.x = w0[0];  b0.y = w1[0];
        b1.x = w0[16]; b1.y = w1[16];
        b2.x = w0[32]; b2.y = w1[32];
        b3.x = w0[48]; b3.y = w1[48];

        acc0 = __builtin_amdgcn_wmma_f32_16x16x4_f32(false, a, false, b0, (short)0, acc0, false, false);
        acc1 = __builtin_amdgcn_wmma_f32_16x16x4_f32(false, a, false, b1, (short)0, acc1, false, false);
        acc2 = __builtin_amdgcn_wmma_f32_16x16x4_f32(false, a, false, b2, (short)0, acc2, false, false);
        acc3 = __builtin_amdgcn_wmma_f32_16x16x4_f32(false, a, false, b3, (short)0, acc3, false, false);
    }

    const float bb0 = bias[j0 + mn];
    const float bb1 = bias[j0 + 16 + mn];
    const float bb2 = bias[j0 + 32 + mn];
    const float bb3 = bias[j0 + 48 + mn];
#pragma unroll
    for (int v = 0; v < 8; ++v) {
        float* crow = C + (size_t)(i0 + v + 8 * half) * HID + j0 + mn;
        if constexpr (NT) {
            __builtin_nontemporal_store(acc0[v] + bb0, crow);
            __builtin_nontemporal_store(acc1[v] + bb1, crow + 16);
            __builtin_nontemporal_store(acc2[v] + bb2, crow + 32);
            __builtin_nontemporal_store(acc3[v] + bb3, crow + 48);
        } else {
            crow[0]  = acc0[v] + bb0;
            crow[16] = acc1[v] + bb1;
            crow[32] = acc2[v] + bb2;
            crow[48] = acc3[v] + bb3;
        }
    }
}

// ---------------------------------------------------------------------------
// Per (head h, 16-row q block): S = Qh @ Kh^T (d=64), eff = S*0.125 +
// (1-mask)*(-1e4); per-row argmax (first occurrence on ties, matching
// jnp.argmax); write the one-hot rows in a single non-temporal v4f pass.
// 8 waves x 16 contiguous k-tiles each (ascending k => '>' preserves
// first-occurrence semantics within a wave).
// ---------------------------------------------------------------------------
__global__ __launch_bounds__(256) void attn_argmax_onehot(
    const float* __restrict__ Q, const float* __restrict__ K,
    const float* __restrict__ mask, float* __restrict__ sampled)
{
    __shared__ float s_val[8][32][8];
    __shared__ int   s_idx[8][32][8];
    __shared__ int   s_amax[16];

    const int h    = blockIdx.y;
    const int q0   = blockIdx.x * 16;           // row within head
    const int lane = threadIdx.x & 31;
    const int wave = threadIdx.x >> 5;
    const int mn   = lane & 15;
    const int half = lane >> 4;

    const float* Qh = Q + (size_t)h * SEQ * HS;
    const float* Kh = K + (size_t)h * SEQ * HS;

    // Preload all 16 A fragments (16 rows x 64 d) into registers.
    v2f afr[16];
    {
        const float* qrow = Qh + (size_t)(q0 + mn) * HS + 2 * half;
#pragma unroll
        for (int d = 0; d < 16; ++d) {
            afr[d].x = qrow[4 * d + 0];
            afr[d].y = qrow[4 * d + 1];
        }
    }

    float best[8];
    int   bidx[8];
#pragma unroll
    for (int v = 0; v < 8; ++v) { best[v] = -3.4e38f; bidx[v] = 0; }

    for (int kt = wave * 16; kt < wave * 16 + 16; ++kt) {
        const int kcol = kt * 16 + mn;          // this lane's k column
        const float* krow = Kh + (size_t)kcol * HS + 2 * half;
        v8f acc = {};
#pragma unroll
        for (int d = 0; d < 16; ++d) {
            v2f b;
            b.x = krow[4 * d + 0];
            b.y = krow[4 * d + 1];
            acc = __builtin_amdgcn_wmma_f32_16x16x4_f32(false, afr[d], false, b, (short)0, acc, false, false);
        }
#pragma unroll
        for (int v = 0; v < 8; ++v) {
            const int qr = q0 + v + 8 * half;   // score row within head
            const float mv  = mask[(size_t)qr * SEQ + kcol];
            const float eff = acc[v] * 0.125f + (1.0f - mv) * -10000.0f;
            if (eff > best[v]) { best[v] = eff; bidx[v] = kcol; }
        }
    }

#pragma unroll
    for (int v = 0; v < 8; ++v) {
        s_val[wave][lane][v] = best[v];
        s_idx[wave][lane][v] = bidx[v];
    }
    __syncthreads();

    // Final per-row reduction: row r lives in lanes [hh*16, hh*16+16) slot v,
    // across all 8 waves. Lowest index wins on exact ties.
    if (threadIdx.x < 16) {
        const int r  = threadIdx.x;
        const int v  = r & 7;
        const int hh = r >> 3;
        float bv = -3.4e38f;
        int   bi = 0x7fffffff;
        for (int w = 0; w < 8; ++w)
            for (int n = 0; n < 16; ++n) {
                const float val = s_val[w][hh * 16 + n][v];
                const int   id  = s_idx[w][hh * 16 + n][v];
                if (val > bv || (val == bv && id < bi)) { bv = val; bi = id; }
            }
        s_amax[r] = bi;
    }
    __syncthreads();

    // One-hot store pass: 16 rows x 2048 cols = 8192 vec4 per workgroup.
    // Non-temporal: this 268 MB stream must not rinse the L2 working set.
    float* out = sampled + ((size_t)h * SEQ + q0) * SEQ;
    for (int p = threadIdx.x; p < (16 * SEQ) / 4; p += 256) {
        const int e = p * 4;
        const int r = e >> 11;                  // e / 2048
        const int k = e & (SEQ - 1);
        const int am = s_amax[r];
        v4f val;
        val.x = (k + 0 == am) ? 1.0f : 0.0f;
        val.y = (k + 1 == am) ? 1.0f : 0.0f;
        val.z = (k + 2 == am) ? 1.0f : 0.0f;
        val.w = (k + 3 == am) ? 1.0f : 0.0f;
        __builtin_nontemporal_store(val, reinterpret_cast<v4f*>(out + (size_t)r * SEQ + k));
    }
}

extern "C" void kernel_launch(void* const* d_in, const int* in_sizes, int n_in,
                              void* d_out, int out_size, void* d_ws, size_t ws_size,
                              hipStream_t stream) {
    const float* q_in = (const float*)d_in[0];
    const float* k_in = (const float*)d_in[1];
    const float* v_in = (const float*)d_in[2];
    const float* mask = (const float*)d_in[3];
    const float* Wq   = (const float*)d_in[4];
    const float* bq   = (const float*)d_in[5];
    const float* Wk   = (const float*)d_in[6];
    const float* bk   = (const float*)d_in[7];
    const float* Wv   = (const float*)d_in[8];
    const float* bv   = (const float*)d_in[9];
    // d_in[10] = is_ingr_to_step (1), d_in[11] = is_gumbel (0): the other
    // branches change the output shape, so out_size pins this path.

    float* out     = (float*)d_out;
    float* sampled = out;                                      // [16,2048,2048]
    float* vout    = out + (size_t)NHEAD * SEQ * SEQ;          // [2048,1024] flat
    float* Qp      = (float*)d_ws;                             // 8 MB
    float* Kp      = Qp + (size_t)SEQ * HID;                   // 8 MB

    proj_gemm_wmma<false><<<256, 256, 0, stream>>>(q_in, Wq, bq, Qp);
    proj_gemm_wmma<false><<<256, 256, 0, stream>>>(k_in, Wk, bk, Kp);
    proj_gemm_wmma<true ><<<256, 256, 0, stream>>>(v_in, Wv, bv, vout);
    attn_argmax_onehot<<<dim3(128, NHEAD), 256, 0, stream>>>(Qp, Kp, mask, sampled);
}